// DeployPostProcessor_47021301957490
// MI455X (gfx1250) — compile-verified
//
#include <hip/hip_runtime.h>
#include <hip/hip_bf16.h>
#include <math.h>

#define NBINS    2048      // 11-bit key prefix histogram
#define CAP      2048      // candidate buffer capacity
#define K_TOP    300
#define NCLS     80
#define NQ       1000
#define NELEM    (NQ * NCLS)   // 80000 per batch
#define NV4      (NELEM / 4)   // 20000 float4 per batch
#define NTHREADS 1024

typedef __attribute__((ext_vector_type(2))) float v2f;
typedef __attribute__((ext_vector_type(8))) float v8f;

// Monotonic unsigned transform of IEEE float bits (order-preserving).
__device__ __forceinline__ unsigned key_of(float x) {
    unsigned b = __float_as_uint(x);
    return (b & 0x80000000u) ? ~b : (b | 0x80000000u);
}
__device__ __forceinline__ float inv_key(unsigned k) {
    unsigned b = (k & 0x80000000u) ? (k & 0x7fffffffu) : ~k;
    return __uint_as_float(b);
}

__global__ __launch_bounds__(NTHREADS)
void detr_topk_kernel(const float* __restrict__ logits,
                      const float* __restrict__ boxes,
                      float* __restrict__ out)
{
    __shared__ unsigned hist[NBINS];
    __shared__ unsigned long long cand[CAP];
    __shared__ int sel_q[((K_TOP + 15) / 16) * 16];  // 304, padded to WMMA group
    __shared__ unsigned s_T, s_cnt;

    const int b   = blockIdx.x;
    const int B   = gridDim.x;
    const int tid = threadIdx.x;
    const float4* lg4 = (const float4*)(logits + (size_t)b * NELEM);

    float* labels_out = out;
    float* boxes_out  = out + (size_t)B * K_TOP;
    float* scores_out = out + (size_t)B * K_TOP * 5;

    // ---- init ----
    hist[tid]        = 0u;
    hist[tid + 1024] = 0u;
    if (tid == 0) s_cnt = 0u;
    __syncthreads();

    // ---- pass 1: histogram of 11-bit key prefix (HBM read, fills L2) ----
    for (int v = tid; v < NV4; v += NTHREADS) {
        float4 f = lg4[v];
        atomicAdd(&hist[key_of(f.x) >> 21], 1u);
        atomicAdd(&hist[key_of(f.y) >> 21], 1u);
        atomicAdd(&hist[key_of(f.z) >> 21], 1u);
        atomicAdd(&hist[key_of(f.w) >> 21], 1u);
    }
    __syncthreads();

    // ---- in-place Hillis-Steele suffix scan: hist[i] = sum_{j>=i} hist0[j] ----
    const int i0 = tid, i1 = tid + 1024;
    for (int off = 1; off < NBINS; off <<= 1) {
        unsigned v0 = (i0 + off < NBINS) ? hist[i0 + off] : 0u;
        unsigned v1 = (i1 + off < NBINS) ? hist[i1 + off] : 0u;
        __syncthreads();
        hist[i0] += v0;
        hist[i1] += v1;
        __syncthreads();
    }
    // threshold bin T = max i with cum[i] >= K_TOP (cum non-increasing -> unique writer)
    {
        unsigned c0 = hist[i0];
        unsigned n0 = (i0 + 1 < NBINS) ? hist[i0 + 1] : 0u;
        if (c0 >= (unsigned)K_TOP && n0 < (unsigned)K_TOP) s_T = (unsigned)i0;
        unsigned c1 = hist[i1];
        unsigned n1 = (i1 + 1 < NBINS) ? hist[i1 + 1] : 0u;
        if (c1 >= (unsigned)K_TOP && n1 < (unsigned)K_TOP) s_T = (unsigned)i1;
    }
    __syncthreads();
    const unsigned T = s_T;

    // ---- pass 2: collect candidates (L2-resident re-read: 320KB/batch << 192MB) ----
    // composite 64-bit key: {key32, ~idx} -> descending sort order == value desc,
    // index asc on ties (matches jax.lax.top_k).
    for (int v = tid; v < NV4; v += NTHREADS) {
        float4 f = lg4[v];
        float vals[4] = {f.x, f.y, f.z, f.w};
        #pragma unroll
        for (int c = 0; c < 4; ++c) {
            unsigned k = key_of(vals[c]);
            if ((k >> 21) >= T) {
                unsigned pos = atomicAdd(&s_cnt, 1u);
                if (pos < CAP) {
                    unsigned idx = (unsigned)(4 * v + c);
                    cand[pos] = ((unsigned long long)k << 32) |
                                (unsigned long long)(~idx);
                }
            }
        }
    }
    __syncthreads();
    const int Cs = (int)(s_cnt < (unsigned)CAP ? s_cnt : (unsigned)CAP);

    // ---- exact ranking: rank(i) = #{ j : cand[j] > cand[i] }  (broadcast LDS reads) ----
    for (int i = tid; i < Cs; i += NTHREADS) {
        unsigned long long comp = cand[i];
        int rank = 0;
        for (int j = 0; j < Cs; ++j)
            rank += (cand[j] > comp) ? 1 : 0;
        if (rank < K_TOP) {
            unsigned idx = ~(unsigned)(comp & 0xffffffffu);
            float x = inv_key((unsigned)(comp >> 32));
            float score = 1.0f / (1.0f + __expf(-x));   // sigmoid only on winners
            int label = (int)(idx % NCLS);
            int q     = (int)(idx / NCLS);
            labels_out[(size_t)b * K_TOP + rank] = (float)label;
            scores_out[(size_t)b * K_TOP + rank] = score;
            sel_q[rank] = q;
        }
    }
    __syncthreads();

    // ---- WMMA box conversion on wave 0: D(16x16) = A(16x4 coeffs) x B(4x16 cxcywh) ----
    // A rows 0..3: [1,0,-.5,0] [0,1,0,-.5] [1,0,.5,0] [0,1,0,.5]; rows 4..15 zero.
    // f32 A/B layout: VGPR selects K low bit, lane half selects K high bit.
    if (tid < 32) {
        const float4* bx4 = (const float4*)(boxes + (size_t)b * NQ * 4);
        const int lane = tid;
        const int m = lane & 15;
        v2f a;
        if (lane < 16) {                       // K=0 (cx), K=1 (cy) coefficients
            a.x = (m == 0 || m == 2) ? 1.0f : 0.0f;
            a.y = (m == 1 || m == 3) ? 1.0f : 0.0f;
        } else {                               // K=2 (w), K=3 (h) coefficients
            a.x = (m == 0) ? -0.5f : ((m == 2) ? 0.5f : 0.0f);
            a.y = (m == 1) ? -0.5f : ((m == 3) ? 0.5f : 0.0f);
        }
        v8f cz = {};
        #pragma unroll 1
        for (int g = 0; g < (K_TOP + 15) / 16; ++g) {
            int r = g * 16 + (lane & 15);
            int q = (r < K_TOP) ? sel_q[r] : 0;        // clamp padded lanes
            float4 bx = bx4[q];                         // gather cxcywh
            v2f bm;
            bm.x = (lane < 16) ? bx.x : bx.z;           // K0=cx / K2=w
            bm.y = (lane < 16) ? bx.y : bx.w;           // K1=cy / K3=h
            // EXEC all-ones here (whole wave active) as WMMA requires.
            v8f d = __builtin_amdgcn_wmma_f32_16x16x4_f32(
                        false, a, false, bm, (short)0, cz, false, false);
            if (lane < 16 && r < K_TOP) {
                // lane n (<16) holds D[0..3][n] in d[0..3] = xyxy of box n
                float4 ov;
                ov.x = d[0]; ov.y = d[1]; ov.z = d[2]; ov.w = d[3];
                *(float4*)(boxes_out + ((size_t)b * K_TOP + r) * 4) = ov;
            }
        }
    }
}

extern "C" void kernel_launch(void* const* d_in, const int* in_sizes, int n_in,
                              void* d_out, int out_size, void* d_ws, size_t ws_size,
                              hipStream_t stream) {
    const float* logits = (const float*)d_in[0];   // [B,1000,80] fp32
    const float* boxes  = (const float*)d_in[1];   // [B,1000,4]  fp32
    float* out = (float*)d_out;                    // labels | boxes | scores, flat
    const int B = in_sizes[0] / NELEM;             // 256
    detr_topk_kernel<<<dim3(B), dim3(NTHREADS), 0, stream>>>(logits, boxes, out);
}